// GATReLU_53197464928901
// MI455X (gfx1250) — compile-verified
//
#include <hip/hip_runtime.h>
#include <math.h>

typedef float v2f __attribute__((ext_vector_type(2)));
typedef float v8f __attribute__((ext_vector_type(8)));

#define NEG_SLOPE 0.2f
#define EPS_SM 1e-16f
#define LDS_STRIDE 132   // pad: bank = (4*col + k) % 64 -> conflict-free b64 reads

// ---------------------------------------------------------------------------
// GEMM: out[M,128] = A[M,128] @ W[128,128] (+bias) (+relu), fp32 WMMA 16x16x4.
// One wave = one 16-row stripe, all 8 column tiles.
// W staged transposed+padded in LDS; all A fragments preloaded to registers.
// ---------------------------------------------------------------------------
__global__ __launch_bounds__(256) void gat_gemm128_wmma(
    const float* __restrict__ A, const float* __restrict__ W,
    const float* __restrict__ bias, float* __restrict__ out,
    int M, int hasBias, int doRelu)
{
    __shared__ float sWt[128 * LDS_STRIDE];   // sWt[col*132 + k] = W[k][col]
    for (int i = threadIdx.x; i < 128 * 128; i += 256) {
        const int k = i >> 7, col = i & 127;
        sWt[col * LDS_STRIDE + k] = W[i];
    }
    __syncthreads();

    const int lane  = threadIdx.x & 31;
    const int wave  = threadIdx.x >> 5;
    const int tileM = blockIdx.x * 8 + wave;          // wave-uniform
    if (tileM * 16 >= M) return;
    const int m0   = tileM * 16;
    const int half = lane >> 4;                       // 0: K+0,1   1: K+2,3
    const int l16  = lane & 15;

    // Preload the entire per-lane A strip: 32 k-steps x 2 floats = 64 VGPRs.
    // 16x4 f32 A layout: row = m0+l16, k offset = half*2.
    const float* Arow = A + (size_t)(m0 + l16) * 128 + half * 2;
    v2f a[32];
    #pragma unroll
    for (int t = 0; t < 32; ++t)
        a[t] = *reinterpret_cast<const v2f*>(Arow + t * 4);

    v8f acc[8];
    #pragma unroll
    for (int c = 0; c < 8; ++c) acc[c] = (v8f){0.f,0.f,0.f,0.f,0.f,0.f,0.f,0.f};

    #pragma unroll
    for (int t = 0; t < 32; ++t) {
        const int krow = t * 4 + half * 2;            // B rows K..K+1 / K+2..K+3
        #pragma unroll
        for (int c = 0; c < 8; ++c) {
            const int col = c * 16 + l16;
            v2f b = *reinterpret_cast<const v2f*>(&sWt[col * LDS_STRIDE + krow]);
            acc[c] = __builtin_amdgcn_wmma_f32_16x16x4_f32(
                false, a[t], false, b, (short)0, acc[c], false, false);
        }
    }

    // D layout: VGPR r -> row m0+r (lanes 0-15) / m0+r+8 (lanes 16-31)
    #pragma unroll
    for (int c = 0; c < 8; ++c) {
        const int col = c * 16 + l16;
        const float bv = hasBias ? bias[col] : 0.0f;
        #pragma unroll
        for (int r = 0; r < 8; ++r) {
            const int row = m0 + r + half * 8;
            float v = acc[c][r] + bv;
            if (doRelu) v = fmaxf(v, 0.0f);
            out[(size_t)row * 128 + col] = v;
        }
    }
}

// ---------------------------------------------------------------------------
// Ordered-uint encoding for monotonic float atomicMax (single u32 atomic).
// ---------------------------------------------------------------------------
__device__ __forceinline__ unsigned fenc(float f) {
    unsigned u = __float_as_uint(f);
    return (u & 0x80000000u) ? ~u : (u | 0x80000000u);
}
__device__ __forceinline__ float fdec(unsigned k) {
    return __uint_as_float((k & 0x80000000u) ? (k ^ 0x80000000u) : ~k);
}
__device__ __forceinline__ float leaky(float v) {
    return v > 0.0f ? v : NEG_SLOPE * v;
}

// ---------------------------------------------------------------------------
// Per-node: alpha dot products (wave reduction), init emax/denom, zero agg buf.
// ---------------------------------------------------------------------------
__global__ __launch_bounds__(256) void gat_node_prep(
    const float* __restrict__ hp, const float* __restrict__ a_src,
    const float* __restrict__ a_dst, float* __restrict__ alphaS,
    float* __restrict__ alphaD, unsigned* __restrict__ emaxEnc,
    float* __restrict__ denom, float* __restrict__ agg, int N)
{
    const int node = (int)((blockIdx.x * 256 + threadIdx.x) >> 5);
    const int lane = threadIdx.x & 31;
    if (node >= N) return;

    float4 h4  = reinterpret_cast<const float4*>(hp + (size_t)node * 128)[lane];
    float4 as4 = reinterpret_cast<const float4*>(a_src)[lane];
    float4 ad4 = reinterpret_cast<const float4*>(a_dst)[lane];
    float ps = h4.x*as4.x + h4.y*as4.y + h4.z*as4.z + h4.w*as4.w;
    float pd = h4.x*ad4.x + h4.y*ad4.y + h4.z*ad4.z + h4.w*ad4.w;
    #pragma unroll
    for (int off = 16; off > 0; off >>= 1) {
        ps += __shfl_down(ps, off, 32);
        pd += __shfl_down(pd, off, 32);
    }
    reinterpret_cast<float4*>(agg + (size_t)node * 128)[lane] =
        make_float4(0.f, 0.f, 0.f, 0.f);
    if (lane == 0) {
        alphaS[node]  = ps;
        alphaD[node]  = pd;
        emaxEnc[node] = fenc(-INFINITY);
        denom[node]   = 0.0f;
    }
}

// ---------------------------------------------------------------------------
// Edge pass 1: segment max of leaky_relu(aS[src]+aD[dst]) over dst.
// Threads e >= E are the implicit self-loops (src = dst = e - E).
// ---------------------------------------------------------------------------
__global__ __launch_bounds__(256) void gat_edge_max(
    const int* __restrict__ src, const int* __restrict__ dst,
    const float* __restrict__ aS, const float* __restrict__ aD,
    unsigned* __restrict__ emaxEnc, int E, int N)
{
    const int e = blockIdx.x * 256 + threadIdx.x;
    if (e >= E + N) return;
    int s, d;
    if (e < E) { s = src[e]; d = dst[e]; } else { s = d = e - E; }
    const float v = leaky(aS[s] + aD[d]);
    atomicMax(&emaxEnc[d], fenc(v));
}

// ---------------------------------------------------------------------------
// Edge pass 2: w = exp(e - emax[dst]); denom[dst] += w; store w per edge.
// ---------------------------------------------------------------------------
__global__ __launch_bounds__(256) void gat_edge_expsum(
    const int* __restrict__ src, const int* __restrict__ dst,
    const float* __restrict__ aS, const float* __restrict__ aD,
    const unsigned* __restrict__ emaxEnc, float* __restrict__ denom,
    float* __restrict__ wEdge, int E, int N)
{
    const int e = blockIdx.x * 256 + threadIdx.x;
    if (e >= E + N) return;
    int s, d;
    if (e < E) { s = src[e]; d = dst[e]; } else { s = d = e - E; }
    const float v = leaky(aS[s] + aD[d]);
    const float w = expf(v - fdec(emaxEnc[d]));
    wEdge[e] = w;
    atomicAdd(&denom[d], w);
}

// ---------------------------------------------------------------------------
// Edge pass 3: agg[dst] += (w/denom[dst]) * h'[src].  One wave per edge,
// lane handles 4 contiguous features (coalesced 512B atomic burst per edge).
// ---------------------------------------------------------------------------
__global__ __launch_bounds__(256) void gat_edge_message(
    const int* __restrict__ src, const int* __restrict__ dst,
    const float* __restrict__ hp, const float* __restrict__ wEdge,
    const float* __restrict__ denom, float* __restrict__ agg, int E, int N)
{
    const int eid  = (int)((blockIdx.x * 256 + threadIdx.x) >> 5);
    const int lane = threadIdx.x & 31;
    if (eid >= E + N) return;
    int s, d;
    if (eid < E) { s = src[eid]; d = dst[eid]; } else { s = d = eid - E; }
    const float alpha = wEdge[eid] / (denom[d] + EPS_SM);
    float4 h4 = reinterpret_cast<const float4*>(hp + (size_t)s * 128)[lane];
    float* o = agg + (size_t)d * 128 + lane * 4;
    atomicAdd(o + 0, alpha * h4.x);
    atomicAdd(o + 1, alpha * h4.y);
    atomicAdd(o + 2, alpha * h4.z);
    atomicAdd(o + 3, alpha * h4.w);
}

// ---------------------------------------------------------------------------
// Elementwise bias + relu over [N,128].
// ---------------------------------------------------------------------------
__global__ __launch_bounds__(256) void gat_bias_relu(
    float* __restrict__ agg, const float* __restrict__ b, int total)
{
    const int i = blockIdx.x * 256 + threadIdx.x;
    if (i >= total) return;
    agg[i] = fmaxf(agg[i] + b[i & 127], 0.0f);
}

// ---------------------------------------------------------------------------
// Pooling: zero, scatter-accumulate, then fused mean + final [64,128]@[128,10].
// ---------------------------------------------------------------------------
__global__ __launch_bounds__(256) void gat_pool_zero(
    float* __restrict__ pooled, float* __restrict__ counts)
{
    const int i = blockIdx.x * 256 + threadIdx.x;
    if (i < 64 * 128) pooled[i] = 0.0f;
    if (i < 64) counts[i] = 0.0f;
}

__global__ __launch_bounds__(256) void gat_pool_acc(
    const float* __restrict__ h, const int* __restrict__ batch,
    float* __restrict__ pooled, float* __restrict__ counts, int N)
{
    const int node = (int)((blockIdx.x * 256 + threadIdx.x) >> 5);
    const int lane = threadIdx.x & 31;
    if (node >= N) return;
    const int g = batch[node];
    float4 h4 = reinterpret_cast<const float4*>(h + (size_t)node * 128)[lane];
    float* p = pooled + (size_t)g * 128 + lane * 4;
    atomicAdd(p + 0, h4.x);
    atomicAdd(p + 1, h4.y);
    atomicAdd(p + 2, h4.z);
    atomicAdd(p + 3, h4.w);
    if (lane == 0) atomicAdd(&counts[g], 1.0f);
}

__global__ __launch_bounds__(640) void gat_final_mlp(
    const float* __restrict__ pooled, const float* __restrict__ counts,
    const float* __restrict__ W1, const float* __restrict__ b1,
    float* __restrict__ out)
{
    const int t = threadIdx.x;
    if (t >= 640) return;
    const int g = t / 10, o = t % 10;
    const float inv = 1.0f / fmaxf(counts[g], 1.0f);
    float acc = b1[o];
    #pragma unroll 8
    for (int k = 0; k < 128; ++k)
        acc += (pooled[g * 128 + k] * inv) * W1[k * 10 + o];
    out[t] = acc;
}

// ---------------------------------------------------------------------------
// Host-side orchestration.
// ---------------------------------------------------------------------------
extern "C" void kernel_launch(void* const* d_in, const int* in_sizes, int n_in,
                              void* d_out, int out_size, void* d_ws, size_t ws_size,
                              hipStream_t stream) {
    const float* x        = (const float*)d_in[0];
    const int*   eidx     = (const int*)  d_in[1];
    // d_in[2] = edge_attr (unused by reference)
    const int*   batch    = (const int*)  d_in[3];
    const float* W0       = (const float*)d_in[4];
    const float* b0       = (const float*)d_in[5];
    const float* Wc       = (const float*)d_in[6];
    const float* att_src  = (const float*)d_in[7];
    const float* att_dst  = (const float*)d_in[8];
    const float* bc       = (const float*)d_in[9];
    const float* W1       = (const float*)d_in[10];
    const float* b1       = (const float*)d_in[11];
    float* outp = (float*)d_out;

    const int N = in_sizes[3];          // 50000 nodes
    const int E = in_sizes[1] / 2;      // 800000 edges
    const int ET = E + N;               // + self loops
    const int* srcIdx = eidx;
    const int* dstIdx = eidx + E;

    // Workspace carve-out
    char* ws = (char*)d_ws;
    size_t off = 0;
    auto carve = [&](size_t bytes) -> char* {
        char* p = ws + off;
        off += (bytes + 255) & ~(size_t)255;
        return p;
    };
    float*    bufA   = (float*)   carve((size_t)N * 128 * 4);
    float*    bufB   = (float*)   carve((size_t)N * 128 * 4);
    float*    aS     = (float*)   carve((size_t)N * 4);
    float*    aD     = (float*)   carve((size_t)N * 4);
    unsigned* emaxE  = (unsigned*)carve((size_t)N * 4);
    float*    denom  = (float*)   carve((size_t)N * 4);
    float*    wEdge  = (float*)   carve((size_t)ET * 4);
    float*    pooled = (float*)   carve(64 * 128 * 4);
    float*    counts = (float*)   carve(64 * 4);
    (void)ws_size; (void)n_in; (void)out_size;

    const int gemmBlocks = ((N / 16) + 7) / 8;          // 8 waves/block, 16 rows/wave
    const int waveNBlocks = (N + 7) / 8;                // wave-per-node kernels
    const int edgeBlocks  = (ET + 255) / 256;           // thread-per-edge kernels
    const int waveEBlocks = (ET + 7) / 8;               // wave-per-edge kernel
    const int ewBlocks    = (N * 128 + 255) / 256;      // elementwise [N,128]

    // h = relu(x @ W0 + b0)
    gat_gemm128_wmma<<<gemmBlocks, 256, 0, stream>>>(x, W0, b0, bufA, N, 1, 1);

    for (int layer = 0; layer < 3; ++layer) {
        const float* Wl  = Wc      + (size_t)layer * 128 * 128;
        const float* asl = att_src + (size_t)layer * 128;
        const float* adl = att_dst + (size_t)layer * 128;
        const float* bcl = bc      + (size_t)layer * 128;

        // h' = h @ Wl (no bias/relu yet)
        gat_gemm128_wmma<<<gemmBlocks, 256, 0, stream>>>(bufA, Wl, nullptr, bufB, N, 0, 0);
        // alpha dots + init emax/denom + zero aggregation target (bufA)
        gat_node_prep<<<waveNBlocks, 256, 0, stream>>>(bufB, asl, adl, aS, aD,
                                                       emaxE, denom, bufA, N);
        gat_edge_max<<<edgeBlocks, 256, 0, stream>>>(srcIdx, dstIdx, aS, aD, emaxE, E, N);
        gat_edge_expsum<<<edgeBlocks, 256, 0, stream>>>(srcIdx, dstIdx, aS, aD,
                                                        emaxE, denom, wEdge, E, N);
        gat_edge_message<<<waveEBlocks, 256, 0, stream>>>(srcIdx, dstIdx, bufB,
                                                          wEdge, denom, bufA, E, N);
        gat_bias_relu<<<ewBlocks, 256, 0, stream>>>(bufA, bcl, N * 128);
    }

    gat_pool_zero<<<(64 * 128 + 255) / 256, 256, 0, stream>>>(pooled, counts);
    gat_pool_acc<<<waveNBlocks, 256, 0, stream>>>(bufA, batch, pooled, counts, N);
    gat_final_mlp<<<1, 640, 0, stream>>>(pooled, counts, W1, b1, outp);
}